// tripletLossDTW_10514079940716
// MI455X (gfx1250) — compile-verified
//
#include <hip/hip_runtime.h>

typedef __attribute__((ext_vector_type(2))) float v2f;
typedef __attribute__((ext_vector_type(8))) float v8f;

#define HH 32
#define WW 32
#define CC 256
#define DD (WW * CC)                   // 8192: row length for both DTW axes
#define IMG ((size_t)HH * WW * CC)     // 262144 elements per batch image
#define TEN ((size_t)32 * IMG)         // 8388608 elements per tensor (anchor/pos/neg)
#define SS 32
#define LPATH 63

#define KC 128                         // K-chunk staged in LDS (contiguous for both axes)
#define LROW 132                       // padded LDS row stride (dwords): 16B-aligned rows,
                                       // bank-conflict-free b64 frag loads (4*row+k distinct mod 64)
#define LMAT (32 * LROW)               // 4224 floats per staged matrix

// ---------------------------------------------------------------------------
// Kernel 1: squared row norms for all 3 tensors, both axes. Branchless:
// DTW row r, element k lives at r*rowStride + (k>>8)*chunkStride + (k&255).
// grid = 3*2*32*32 = 6144 blocks, 256 threads.
// ---------------------------------------------------------------------------
__global__ void dtw_row_norm_kernel(const float* __restrict__ feat,
                                    float* __restrict__ norms) {
  int id = blockIdx.x;
  int t = id >> 11, axis = (id >> 10) & 1, b = (id >> 5) & 31, row = id & 31;
  size_t rowStride   = axis ? (size_t)CC : (size_t)DD;
  size_t chunkStride = axis ? (size_t)DD : (size_t)CC;
  const float* rowp = feat + (size_t)t * TEN + (size_t)b * IMG + (size_t)row * rowStride;

  float s = 0.0f;
  for (int k = threadIdx.x * 4; k < DD; k += blockDim.x * 4) {
    const float4 v = *(const float4*)(rowp + (size_t)(k >> 8) * chunkStride + (k & 255));
    s += v.x * v.x + v.y * v.y + v.z * v.z + v.w * v.w;
  }
  __shared__ float red[256];
  red[threadIdx.x] = s;
  __syncthreads();
  for (int off = 128; off > 0; off >>= 1) {
    if (threadIdx.x < off) red[threadIdx.x] += red[threadIdx.x + off];
    __syncthreads();
  }
  if (threadIdx.x == 0) norms[id] = red[0];
}

// ---------------------------------------------------------------------------
// Kernel 2: 32x32 cost matrices via V_WMMA_F32_16X16X4_F32, LDS-staged.
// grid = 128 blocks (pair*64 + axis*32 + b), 128 threads = 4 waves, each wave
// owns one 16x16 tile. Per 128-wide K chunk: cooperative float4 global->LDS
// stage (each element read once per block), prefetch of the next chunk, then
// 32 fully-unrolled WMMA steps with ds_load_b64 fragment reads.
// ---------------------------------------------------------------------------
__global__ void dtw_cost_wmma_kernel(const float* __restrict__ feat,
                                     const float* __restrict__ norms,
                                     float* __restrict__ cost) {
  int id = blockIdx.x;
  int pair = id >> 6, axis = (id >> 5) & 1, b = id & 31;

  size_t rowStride = axis ? (size_t)CC : (size_t)DD;
  const float* Abase = feat + (size_t)b * IMG;                        // anchor
  const float* Bbase = feat + (size_t)(1 + pair) * TEN + (size_t)b * IMG;

  __shared__ float lds[2 * LMAT];                                     // As | Bs

  int tid  = threadIdx.x;
  int wave = tid >> 5;
  int lane = tid & 31;
  int ti = (wave >> 1) * 16;   // output row tile
  int tj = (wave & 1) * 16;    // output col tile
  int ln = lane & 15, hi = lane >> 4;

  // LDS fragment base offsets (dwords) for this lane.
  int la = (ti + ln) * LROW + 2 * hi;          // A frag: row ti+ln, K pair 2*hi
  int lb = LMAT + (tj + ln) * LROW + 2 * hi;   // B frag: row tj+ln

  // Loader mapping: 1024 float4 per matrix / 128 threads = 8 each.
  int lrow0 = tid >> 2;                        // 0..31 (with e offset below)
  // Prefetch mapping: 128 threads cover 32 rows x 4 x 128B segments.
  int prow = tid >> 2;
  int pcol = (tid & 3) << 5;                   // 0,32,64,96 floats

  v8f acc = {};

  for (int kc = 0; kc < DD; kc += KC) {
    // chunkOff: uniform scalar; chunk is contiguous per row for both axes.
    size_t chunkOff = axis ? ((size_t)(kc >> 8) * DD + (size_t)(kc & 255))
                           : (size_t)kc;

    __syncthreads();   // previous chunk fully consumed before overwrite
#pragma unroll
    for (int e = 0; e < 8; ++e) {
      int idx = e * 128 + tid;                 // 0..1023
      int row = idx >> 5;
      int c4  = (idx & 31) << 2;               // float offset within chunk
      const float* ga = Abase + (size_t)row * rowStride + chunkOff + c4;
      const float* gb = Bbase + (size_t)row * rowStride + chunkOff + c4;
      *(float4*)&lds[row * LROW + c4]        = *(const float4*)ga;
      *(float4*)&lds[LMAT + row * LROW + c4] = *(const float4*)gb;
    }
    __syncthreads();   // staged data visible to all 4 waves

    // Prefetch next chunk into cache while this chunk's WMMAs run.
    if (kc + KC < DD) {
      int kn = kc + KC;
      size_t nOff = axis ? ((size_t)(kn >> 8) * DD + (size_t)(kn & 255))
                         : (size_t)kn;
      __builtin_prefetch(Abase + (size_t)prow * rowStride + nOff + pcol, 0, 0);
      __builtin_prefetch(Bbase + (size_t)prow * rowStride + nOff + pcol, 0, 0);
    }

    // 32 WMMA steps over this chunk (K=4 each), fully unrolled.
#pragma unroll
    for (int kk = 0; kk < KC; kk += 4) {
      v2f av = *(const v2f*)&lds[la + kk];
      v2f bv = *(const v2f*)&lds[lb + kk];
      acc = __builtin_amdgcn_wmma_f32_16x16x4_f32(false, av, false, bv,
                                                  (short)0, acc, false, false);
    }
  }

  float div = (axis == 0) ? 8.0f : 1.0f;
  const float* aa = norms + ((0 * 2 + axis) * 32 + b) * 32;          // anchor norms
  const float* bb = norms + (((1 + pair) * 2 + axis) * 32 + b) * 32; // other norms
  float* out = cost + (size_t)id * (SS * SS);

#pragma unroll
  for (int r = 0; r < 8; ++r) {
    int M = ti + r + 8 * hi;   // C/D layout: VGPR r -> M=r (lanes 0-15), M=r+8 (16-31)
    int N = tj + ln;
    float d2 = aa[M] + bb[N] - 2.0f * acc[r];
    d2 = fmaxf(d2, 0.0f);
    out[M * SS + N] = sqrtf(d2 / div) + 1e-8f;
  }
}

// ---------------------------------------------------------------------------
// Kernel 3: DTW backpointers (anti-diagonal wavefront, one wave per problem)
// followed by lane-0 backtrack. grid = 128 blocks, 32 threads.
// paths[pid] = [ii(63) | jj(63) | mask(63)] ints.
// ---------------------------------------------------------------------------
__global__ void dtw_wavefront_kernel(const float* __restrict__ cost,
                                     int* __restrict__ paths) {
  int pid = blockIdx.x;
  const float* m = cost + (size_t)pid * (SS * SS);
  __shared__ int ptr[SS * SS];

  int lane = threadIdx.x;      // = column j
  float pc = 0.f, pl = 0.f;    // diagonal d-1: cell (d-1-j, j)
  float p2c = 0.f, p2l = 0.f;  // diagonal d-2

  for (int d = 0; d < 2 * SS - 1; ++d) {
    int i = d - lane, j = lane;
    float upc = pc, upl = pl;                 // (i-1, j)
    float lc = __shfl_up(pc, 1);              // (i,   j-1)
    float ll = __shfl_up(pl, 1);
    float dc = __shfl_up(p2c, 1);             // (i-1, j-1)
    float dl = __shfl_up(p2l, 1);

    float c = 0.f, l = 0.f;
    if (i >= 0 && i < SS) {
      float mm = m[i * SS + j];
      int p;
      if (i == 0 && j == 0) { c = mm; l = 1.f; p = 3; }
      else if (i == 0)      { c = lc + mm;  l = ll + 1.f;  p = 1; }
      else if (j == 0)      { c = upc + mm; l = upl + 1.f; p = 0; }
      else {
        float a0 = upc / upl, a1 = lc / ll, a2 = dc / dl;
        p = 0; float best = a0;               // tie order: up, left, diag
        if (a1 < best) { best = a1; p = 1; }
        if (a2 < best) { best = a2; p = 2; }
        float bc = (p == 0) ? upc : ((p == 1) ? lc : dc);
        float bl = (p == 0) ? upl : ((p == 1) ? ll : dl);
        c = mm + bc; l = bl + 1.f;
      }
      ptr[i * SS + j] = p;
    }
    p2c = pc; p2l = pl;
    pc = c;  pl = l;
  }
  __syncthreads();

  if (lane == 0) {
    int i = SS - 1, j = SS - 1, act = 1;
    int* out = paths + (size_t)pid * (3 * LPATH);
    for (int t = 0; t < LPATH; ++t) {
      out[t] = i; out[LPATH + t] = j; out[2 * LPATH + t] = act;
      int p = ptr[i * SS + j];
      int halt = (p == 3) || (!act);
      int di = (p == 0 || p == 2) ? 1 : 0;
      int dj = (p == 1 || p == 2) ? 1 : 0;
      i = halt ? i : i - di;
      j = halt ? j : j - dj;
      act = act && (p != 3);
    }
  }
}

// ---------------------------------------------------------------------------
// Kernel 4: masked aligned distance. grid = 64 blocks (q*32 + b), 256 threads.
// dist[q*32+b] = sum over active (px,py) of sqrt(sum_c (v1-v2)^2) / count.
// ---------------------------------------------------------------------------
__global__ void dtw_aligned_dist_kernel(const float* __restrict__ feat,
                                        const int* __restrict__ paths,
                                        float* __restrict__ dist) {
  int q = blockIdx.x >> 5, b = blockIdx.x & 31;
  const float* A = feat + (size_t)b * IMG;                          // anchor
  const float* O = feat + (size_t)(1 + q) * TEN + (size_t)b * IMG;  // pos/neg
  const int* px = paths + (size_t)((q * 2 + 0) * 32 + b) * (3 * LPATH);
  const int* py = paths + (size_t)((q * 2 + 1) * 32 + b) * (3 * LPATH);

  float s = 0.0f;
  int cnt = 0;
  for (int t = threadIdx.x; t < LPATH * LPATH; t += blockDim.x) {
    int ix = t / LPATH, iy = t - ix * LPATH;
    if (px[2 * LPATH + ix] && py[2 * LPATH + iy]) {
      int x1 = px[ix], x2 = px[LPATH + ix];
      int y1 = py[iy], y2 = py[LPATH + iy];
      const float4* v1 = (const float4*)(A + (size_t)x1 * DD + (size_t)y1 * CC);
      const float4* v2 = (const float4*)(O + (size_t)x2 * DD + (size_t)y2 * CC);
      float acc = 0.0f;
#pragma unroll 4
      for (int c = 0; c < CC / 4; ++c) {
        float4 u = v1[c], w = v2[c];
        float dx = u.x - w.x, dy = u.y - w.y, dz = u.z - w.z, dw = u.w - w.w;
        acc += dx * dx + dy * dy + dz * dz + dw * dw;
      }
      s += sqrtf(acc);
      cnt++;
    }
  }
  __shared__ float rs[256];
  __shared__ int rc[256];
  rs[threadIdx.x] = s; rc[threadIdx.x] = cnt;
  __syncthreads();
  for (int off = 128; off > 0; off >>= 1) {
    if (threadIdx.x < off) {
      rs[threadIdx.x] += rs[threadIdx.x + off];
      rc[threadIdx.x] += rc[threadIdx.x + off];
    }
    __syncthreads();
  }
  if (threadIdx.x == 0) dist[q * 32 + b] = rs[0] / (float)rc[0];
}

// ---------------------------------------------------------------------------
// Kernel 5: triplet hinge reduction. 1 block, 32 threads (one wave).
// ---------------------------------------------------------------------------
__global__ void dtw_triplet_loss_kernel(const float* __restrict__ dist,
                                        float* __restrict__ out) {
  int b = threadIdx.x;
  float v = fmaxf(dist[b] - dist[32 + b] + 0.1f, 0.0f);
  for (int off = 16; off > 0; off >>= 1) v += __shfl_down(v, off);
  if (b == 0) out[0] = v;
}

// ---------------------------------------------------------------------------
extern "C" void kernel_launch(void* const* d_in, const int* in_sizes, int n_in,
                              void* d_out, int out_size, void* d_ws, size_t ws_size,
                              hipStream_t stream) {
  const float* feat = (const float*)d_in[0];  // (3, 32, 32, 32, 256) f32
  float* out = (float*)d_out;

  char* ws = (char*)d_ws;
  float* norms = (float*)ws;                                  //  6144 floats
  float* cost  = (float*)(ws + 24576);                        // 128*1024 floats
  int*   paths = (int*)  (ws + 24576 + 524288);               // 128*189 ints
  float* dist  = (float*)(ws + 24576 + 524288 + 96768);       // 64 floats

  dtw_row_norm_kernel    <<<6144, 256, 0, stream>>>(feat, norms);
  dtw_cost_wmma_kernel   <<<128, 128, 0, stream>>>(feat, norms, cost);
  dtw_wavefront_kernel   <<<128, 32, 0, stream>>>(cost, paths);
  dtw_aligned_dist_kernel<<<64, 256, 0, stream>>>(feat, paths, dist);
  dtw_triplet_loss_kernel<<<1, 32, 0, stream>>>(dist, out);
}